// Quantiser_54949811585515
// MI455X (gfx1250) — compile-verified
//
#include <hip/hip_runtime.h>

// Problem constants (fixed by the reference setup_inputs):
//   quant_input: (32, 64, 64, 64) f32   codebook: (512, 64) f32
#define NB      32
#define NC      64
#define HW      4096            // 64*64
#define SPROWS  131072          // B*H*W spatial rows
#define KCODES  512
#define DDIM    64
#define NELEM   8388608         // B*C*H*W

typedef __attribute__((ext_vector_type(2))) float v2f;
typedef __attribute__((ext_vector_type(8))) float v8f;

// ---------------------------------------------------------------------------
// Kernel 1: fp32 WMMA GEMM (x · codebook^T) + fused argmax over 512 codes.
//
// WMMA f32 16x16x4 fragment layouts (wave32):
//   A 16x4 : row m = lane&15; half = lane>>4 selects K {+0,+1} vs {+2,+3}
//   B 4x16 : col n = lane&15; VGPR0 = K rows {0,2}, VGPR1 = {1,3}
//   C  16x16: col n = lane&15; VGPR v = row v + 8*half
//
// The codebook is staged in LDS pre-swizzled into B-fragment order:
//   sBsw[((t*16 + s)*2 + half)*32 + n*2 + j]  (t = N-tile, s = K-step, j = 0/1)
// so a lane's per-step pair is one contiguous ds_load_b64 with an immediate
// offset (s*256 B) from a single per-tile base, and the 32 lanes' 64 dwords
// map onto all 64 LDS banks exactly once (conflict-free).
// B fragments are double-buffered across N-tiles: tile t+1's 16 b64 loads are
// issued before tile t's 32-WMMA chain, so waits become dscnt<=16, not 0.
// ---------------------------------------------------------------------------
__global__ __launch_bounds__(256) void vq_argmin_wmma(
    const float* __restrict__ x, const float* __restrict__ cb,
    int* __restrict__ idx_out)
{
    extern __shared__ float smem[];
    float* sBsw = smem;                       // [32768] swizzled codebook
    float* hesq = smem + KCODES * DDIM;       // [512]   0.5*||e_k||^2

    // Stage codebook into B-fragment order (coalesced global reads).
    for (int g = threadIdx.x; g < KCODES * DDIM; g += blockDim.x) {
        const int n  = g >> 6, c = g & 63;       // code row, channel
        const int t  = n >> 4, nl = n & 15;
        const int s  = c >> 2;
        const int hf = (c >> 1) & 1, j = c & 1;
        sBsw[((t * 16 + s) * 2 + hf) * 32 + nl * 2 + j] = cb[g];
    }
    for (int n = threadIdx.x; n < KCODES; n += blockDim.x) {
        float s = 0.f;
        for (int c = 0; c < DDIM; ++c) { const float v = cb[n * DDIM + c]; s += v * v; }
        hesq[n] = 0.5f * s;
    }
    __syncthreads();

    const int lane = threadIdx.x & 31;
    const int wave = threadIdx.x >> 5;
    const int half = lane >> 4;
    const int nl   = lane & 15;
    const int m    = lane & 15;

    // Per-lane base into the swizzled B region: tile stride = 1024 floats.
    const float* pB = sBsw + half * 32 + nl * 2;

    // Block covers 512 rows; 8 waves x 2 iterations x 32 rows (2 M-tiles).
    for (int it = 0; it < 2; ++it) {
        const int r0   = blockIdx.x * 512 + it * 256 + wave * 32;
        const int base = (r0 >> 12) * 262144 + (r0 & 4095);

        // Preload A fragments for both 16-row tiles (channel stride = 4096).
        v2f a0v[16], a1v[16];
        #pragma unroll
        for (int s = 0; s < 16; ++s) {
            const int c0 = 4 * s + 2 * half;
            a0v[s].x = x[base + m      + (c0    ) * 4096];
            a0v[s].y = x[base + m      + (c0 + 1) * 4096];
            a1v[s].x = x[base + 16 + m + (c0    ) * 4096];
            a1v[s].y = x[base + 16 + m + (c0 + 1) * 4096];
        }

        float bs0[8], bs1[8];
        int   bi0[8], bi1[8];
        #pragma unroll
        for (int v = 0; v < 8; ++v) {
            bs0[v] = -3.4e38f; bs1[v] = -3.4e38f; bi0[v] = 0; bi1[v] = 0;
        }

        auto loadB = [&](v2f* bf, int t) {
            const float* p = pB + t * 1024;
            #pragma unroll
            for (int s = 0; s < 16; ++s)
                bf[s] = *(const v2f*)(p + s * 64);     // ds_load_b64, imm offset
        };
        auto mmaTile = [&](const v2f* bf, int t) {
            v8f acc0 = {}, acc1 = {};
            #pragma unroll
            for (int s = 0; s < 16; ++s) {
                acc0 = __builtin_amdgcn_wmma_f32_16x16x4_f32(
                           false, a0v[s], false, bf[s], (short)0, acc0, false, false);
                acc1 = __builtin_amdgcn_wmma_f32_16x16x4_f32(
                           false, a1v[s], false, bf[s], (short)0, acc1, false, false);
            }
            const float hq    = hesq[t * 16 + nl];
            const int   nglob = t * 16 + nl;
            #pragma unroll
            for (int v = 0; v < 8; ++v) {
                const float s0 = acc0[v] - hq;   // argmax of x.e - 0.5|e|^2
                if (s0 > bs0[v]) { bs0[v] = s0; bi0[v] = nglob; }
                const float s1 = acc1[v] - hq;
                if (s1 > bs1[v]) { bs1[v] = s1; bi1[v] = nglob; }
            }
        };

        // Software-pipelined N-tile loop, unrolled by 2 so both B buffers
        // stay register-resident.
        v2f bf0[16], bf1[16];
        loadB(bf0, 0);
        for (int tt = 0; tt < 16; ++tt) {
            const int t0 = 2 * tt;
            loadB(bf1, t0 + 1);          // prefetch while chain 0 runs
            mmaTile(bf0, t0);
            if (tt < 15) loadB(bf0, t0 + 2);
            mmaTile(bf1, t0 + 1);
        }

        // Reduce over the 16 lanes holding each row (xor masks 8..1 never
        // cross the lane-16 boundary). Ties -> lower index (jnp.argmin).
        #pragma unroll
        for (int v = 0; v < 8; ++v) {
            float s = bs0[v]; int i = bi0[v];
            #pragma unroll
            for (int off = 8; off >= 1; off >>= 1) {
                const float os = __shfl_xor(s, off, 32);
                const int   oi = __shfl_xor(i, off, 32);
                if (os > s || (os == s && oi < i)) { s = os; i = oi; }
            }
            if (nl == 0) idx_out[r0 + v + 8 * half] = i;

            float s2 = bs1[v]; int i2 = bi1[v];
            #pragma unroll
            for (int off = 8; off >= 1; off >>= 1) {
                const float os = __shfl_xor(s2, off, 32);
                const int   oi = __shfl_xor(i2, off, 32);
                if (os > s2 || (os == s2 && oi < i2)) { s2 = os; i2 = oi; }
            }
            if (nl == 0) idx_out[r0 + 16 + v + 8 * half] = i2;
        }
    }
}

// ---------------------------------------------------------------------------
// Kernel 2: gather codebook rows into (B,C,H,W), STE arithmetic x + (q - x),
// fused 1.2*MSE loss reduction, and index->float cast for the output tail.
// ---------------------------------------------------------------------------
__global__ __launch_bounds__(256) void vq_gather_loss(
    const float* __restrict__ x, const float* __restrict__ cb,
    const int* __restrict__ idx, float* __restrict__ out,
    float* __restrict__ loss, float* __restrict__ idxf)
{
    const int tid = blockIdx.x * blockDim.x + threadIdx.x;
    const int nth = gridDim.x * blockDim.x;

    float lsum = 0.f;
    for (int e = tid; e < NELEM; e += nth) {
        const int b   = e >> 18;            // / (C*H*W)
        const int rem = e & 262143;
        const int c   = rem >> 12;          // channel
        const int sp  = rem & 4095;         // h*W + w
        const int r   = (b << 12) + sp;     // spatial row index
        const float q  = cb[idx[r] * DDIM + c];
        const float xv = x[e];
        const float d  = q - xv;
        out[e] = xv + d;                    // straight-through estimator
        lsum += d * d;
    }
    for (int t = tid; t < SPROWS; t += nth) idxf[t] = (float)idx[t];

    // Block reduction of the squared-error partial sums.
    #pragma unroll
    for (int off = 16; off >= 1; off >>= 1) lsum += __shfl_xor(lsum, off, 32);
    __shared__ float wsum[8];
    const int lane = threadIdx.x & 31, wv = threadIdx.x >> 5;
    if (lane == 0) wsum[wv] = lsum;
    __syncthreads();
    if (threadIdx.x == 0) {
        float s = 0.f;
        for (int i = 0; i < 8; ++i) s += wsum[i];
        // loss = codebook_loss + 0.2*commitment = 1.2 * mean((q-x)^2)
        atomicAdd(loss, s * (1.2f / (float)NELEM));
    }
}

extern "C" void kernel_launch(void* const* d_in, const int* in_sizes, int n_in,
                              void* d_out, int out_size, void* d_ws, size_t ws_size,
                              hipStream_t stream) {
    const float* x  = (const float*)d_in[0];   // (32,64,64,64) f32
    const float* cb = (const float*)d_in[1];   // (512,64) f32

    float* out  = (float*)d_out;               // [0, NELEM)        quant_out
    float* loss = out + NELEM;                 // [NELEM]           loss scalar
    float* idxf = out + NELEM + 1;             // [NELEM+1, +SPROWS) indices
    int*   idx  = (int*)d_ws;                  // scratch: 131072 int32

    const size_t lds_bytes = (size_t)KCODES * DDIM * sizeof(float)
                           + (size_t)KCODES * sizeof(float);   // 130 KB

    // 256 blocks x 512 rows = 131072 spatial rows.
    vq_argmin_wmma<<<256, 256, lds_bytes, stream>>>(x, cb, idx);

    hipMemsetAsync(loss, 0, sizeof(float), stream);   // graph-capturable
    vq_gather_loss<<<512, 256, 0, stream>>>(x, cb, idx, out, loss, idxf);
}